// AEEncoder_10926396801077
// MI455X (gfx1250) — compile-verified
//
#include <hip/hip_runtime.h>
#include <hip/hip_bf16.h>

// ---------------------------------------------------------------------------
// AEEncoder forward for MI455X (gfx1250).
// Layer 1 is 1024 independent [256x64]x[64x8] GEMMs -> v_wmma_f32_16x16x4_f32
// (f32 matrix pipe; problem is L2-resident and latency bound, so we keep full
// f32 precision instead of downcasting).
// ---------------------------------------------------------------------------

typedef __attribute__((ext_vector_type(2))) float v2f;
typedef __attribute__((ext_vector_type(8))) float v8f;

#define B_    256
#define G_    20000
#define N_TF  1024
#define NPT   8      // nodes per TF
#define GPT   64     // genes per TF
#define H_    (N_TF * NPT)   // 8192
#define EPS_  1e-5f
#define SLOPE 0.01f

#ifndef HAVE_WMMA_F32
#if defined(__has_builtin)
#if __has_builtin(__builtin_amdgcn_wmma_f32_16x16x4_f32)
#define HAVE_WMMA_F32 1
#endif
#endif
#endif

static __device__ inline v8f wmma_f32_16x16x4(v2f a, v2f b, v8f c) {
#if defined(HAVE_WMMA_F32)
  // 8 args: (neg_a, A, neg_b, B, c_mod, C, reuse_a, reuse_b)
  return __builtin_amdgcn_wmma_f32_16x16x4_f32(
      false, a, false, b, (short)0, c, false, false);
#else
  v8f d;
  asm volatile("v_wmma_f32_16x16x4_f32 %0, %1, %2, %3"
               : "=v"(d)
               : "v"(a), "v"(b), "v"(c));
  return d;
#endif
}

// --------------------------------------------------------------------------
// Kernel 1: per-TF GEMM  h[b, t*8+n] = sum_j w1[(t*8+n)*64+j] * x[b, g[t,j]]
// One workgroup per TF, 8 waves. LDS rows padded to 68 floats (stride 68
// dwords => A-fragment float2 reads land on 32 distinct banks, no conflicts;
// stride 64 would be a 16-way conflict on the 64-bank LDS).
// --------------------------------------------------------------------------
__global__ __launch_bounds__(256) void l1_wmma_kernel(
    const float* __restrict__ x, const float* __restrict__ w1,
    const int* __restrict__ in_idx1, float* __restrict__ h) {
  __shared__ float lds_x[128][68];  // 128 batch rows x 64 genes (padded)
  __shared__ float lds_w[16][68];   // 16 node rows (8 real + 8 zero) x 64 K
  __shared__ int   lds_g[GPT];

  const int t   = blockIdx.x;
  const int tid = threadIdx.x;

  // gene indices for this TF (node 0's edges; all 8 nodes share them)
  if (tid < GPT) lds_g[tid] = in_idx1[(size_t)t * (NPT * GPT) + tid];

  // weights: B-matrix rows n=0..7 real, n=8..15 zero (exact zero padding)
  if (tid < 128) {
    const int n  = tid >> 4;          // 0..7
    const int j0 = (tid & 15) * 4;    // 0..60
    const float* wp = w1 + (size_t)(t * NPT + n) * GPT + j0;
#pragma unroll
    for (int q = 0; q < 4; ++q) lds_w[n][j0 + q] = wp[q];
  } else {
    const int n  = ((tid - 128) >> 4) + 8;  // 8..15
    const int j0 = (tid & 15) * 4;
#pragma unroll
    for (int q = 0; q < 4; ++q) lds_w[n][j0 + q] = 0.0f;
  }

  const int lane = tid & 31;
  const int wave = tid >> 5;          // 0..7 -> one 16-row batch tile each
  const int m    = lane & 15;         // A/B column-lane id
  const int koff = (lane >> 4) * 2;   // lanes 16..31 carry K=2,3

  for (int chunk = 0; chunk < 2; ++chunk) {
    __syncthreads();  // also protects lds_x reuse across chunks
    {
      // stage gathered x columns: each thread fills half a batch row
      const int row = tid >> 1;                // 0..127
      const int j0  = (tid & 1) * 32;
      const float* xr = x + (size_t)(chunk * 128 + row) * G_;
#pragma unroll 8
      for (int j = 0; j < 32; ++j) lds_x[row][j0 + j] = xr[lds_g[j0 + j]];
    }
    __syncthreads();

    v8f acc = {};
#pragma unroll
    for (int step = 0; step < 16; ++step) {
      const int kb = step * 4 + koff;
      v2f a, b;
      a.x = lds_x[wave * 16 + m][kb];      // A: 16x4 f32 tile (batch x K)
      a.y = lds_x[wave * 16 + m][kb + 1];
      b.x = lds_w[m][kb];                  // B: 4x16 f32 tile (K x nodes)
      b.y = lds_w[m][kb + 1];
      acc = wmma_f32_16x16x4(a, b, acc);
    }

    // D layout: vgpr r, lane l -> M = r + 8*(l>=16), N = l&15. Keep N<8.
    const int N = lane & 15;
    if (N < NPT) {
      const int Mbase = (lane >> 4) * 8;
      const size_t col = (size_t)t * NPT + N;
#pragma unroll
      for (int r = 0; r < 8; ++r) {
        const int M = chunk * 128 + wave * 16 + Mbase + r;
        h[(size_t)M * H_ + col] = acc[r];
      }
    }
  }
}

// --------------------------------------------------------------------------
// Kernel 2/4: per-column batch stats (torch BN training semantics: biased
// variance). One thread per column; reads are coalesced across threads.
// --------------------------------------------------------------------------
__global__ __launch_bounds__(256) void bn_stats_kernel(
    const float* __restrict__ v, int ncols, float* __restrict__ mean,
    float* __restrict__ istd) {
  const int o = blockIdx.x * blockDim.x + threadIdx.x;
  if (o >= ncols) return;
  float s = 0.0f, s2 = 0.0f;
  for (int b = 0; b < B_; ++b) {
    const float u = v[(size_t)b * ncols + o];
    s += u;
    s2 += u * u;
  }
  const float mu  = s * (1.0f / B_);
  const float var = s2 * (1.0f / B_) - mu * mu;
  mean[o] = mu;
  istd[o] = rsqrtf(var + EPS_);
}

// --------------------------------------------------------------------------
// Kernel 3: fused BN(h) + LeakyReLU + block-diagonal layer 2.
// z[b,t] = sum_{n<8} w2[t*8+n] * lrelu((h[b,t*8+n]-mu)*istd)
// --------------------------------------------------------------------------
__global__ __launch_bounds__(256) void l2_fused_kernel(
    const float* __restrict__ h, const float* __restrict__ mean1,
    const float* __restrict__ istd1, const float* __restrict__ w2,
    float* __restrict__ z) {
  const int idx = blockIdx.x * blockDim.x + threadIdx.x;  // b*N_TF + t
  if (idx >= B_ * N_TF) return;
  const int b = idx >> 10;
  const int t = idx & (N_TF - 1);
  float acc = 0.0f;
#pragma unroll
  for (int n = 0; n < NPT; ++n) {
    const int o = t * NPT + n;
    float u = (h[(size_t)b * H_ + o] - mean1[o]) * istd1[o];
    u = (u > 0.0f) ? u : SLOPE * u;
    acc += w2[o] * u;
  }
  z[idx] = acc;
}

// --------------------------------------------------------------------------
// Kernel 5: final BN(z) + LeakyReLU -> output [256, 1024]
// --------------------------------------------------------------------------
__global__ __launch_bounds__(256) void norm_out_kernel(
    const float* __restrict__ z, const float* __restrict__ mean2,
    const float* __restrict__ istd2, float* __restrict__ out) {
  const int idx = blockIdx.x * blockDim.x + threadIdx.x;
  if (idx >= B_ * N_TF) return;
  const int t = idx & (N_TF - 1);
  float u = (z[idx] - mean2[t]) * istd2[t];
  out[idx] = (u > 0.0f) ? u : SLOPE * u;
}

// --------------------------------------------------------------------------
// Launch
// --------------------------------------------------------------------------
extern "C" void kernel_launch(void* const* d_in, const int* in_sizes, int n_in,
                              void* d_out, int out_size, void* d_ws,
                              size_t ws_size, hipStream_t stream) {
  (void)in_sizes; (void)n_in; (void)out_size; (void)ws_size;

  const float* x       = (const float*)d_in[0];
  const float* w1      = (const float*)d_in[1];
  const int*   in_idx1 = (const int*)d_in[2];
  // d_in[3] out_idx1, d_in[5] in_idx2, d_in[6] out_idx2 are implied by layout
  const float* w2      = (const float*)d_in[4];
  float* out           = (float*)d_out;

  // workspace layout
  char* ws = (char*)d_ws;
  float* h      = (float*)(ws);                                  // 8 MB
  float* mean1  = (float*)(ws + (size_t)B_ * H_ * 4);            // 32 KB
  float* istd1  = (float*)(ws + (size_t)B_ * H_ * 4 + H_ * 4);   // 32 KB
  float* z      = (float*)(ws + (size_t)B_ * H_ * 4 + 2 * H_ * 4);  // 1 MB
  float* mean2  = (float*)((char*)z + (size_t)B_ * N_TF * 4);
  float* istd2  = (float*)((char*)mean2 + N_TF * 4);

  l1_wmma_kernel<<<N_TF, 256, 0, stream>>>(x, w1, in_idx1, h);
  bn_stats_kernel<<<H_ / 256, 256, 0, stream>>>(h, H_, mean1, istd1);
  l2_fused_kernel<<<(B_ * N_TF) / 256, 256, 0, stream>>>(h, mean1, istd1, w2, z);
  bn_stats_kernel<<<N_TF / 256, 256, 0, stream>>>(z, N_TF, mean2, istd2);
  norm_out_kernel<<<(B_ * N_TF) / 256, 256, 0, stream>>>(z, mean2, istd2, out);
}